// GCN_5626407158020
// MI455X (gfx1250) — compile-verified
//
#include <hip/hip_runtime.h>
#include <hip/hip_bf16.h>
#include <math.h>

typedef float v2f __attribute__((ext_vector_type(2)));
typedef float v4f __attribute__((ext_vector_type(4)));
typedef float v8f __attribute__((ext_vector_type(8)));

// ---------------- utility ----------------
__global__ void zero32_kernel(unsigned int* __restrict__ p, int n) {
    int i = blockIdx.x * blockDim.x + threadIdx.x;
    if (i < n) p[i] = 0u;
}

// ---------------- degrees & norms ----------------
__global__ void deg_kernel(const int* __restrict__ src, const int* __restrict__ dst,
                           int* __restrict__ odeg, int* __restrict__ ideg, int E) {
    int e = blockIdx.x * blockDim.x + threadIdx.x;
    if (e >= E) return;
    atomicAdd(&odeg[src[e]], 1);
    atomicAdd(&ideg[dst[e]], 1);
}

__global__ void norm_kernel(const int* __restrict__ odeg, const int* __restrict__ ideg,
                            float* __restrict__ onorm, float* __restrict__ inorm, int N) {
    int i = blockIdx.x * blockDim.x + threadIdx.x;
    if (i >= N) return;
    onorm[i] = rsqrtf(fmaxf((float)odeg[i], 1.0f));
    inorm[i] = rsqrtf(fmaxf((float)ideg[i], 1.0f));
}

// ---------------- exclusive scan of in-degree -> rowptr ----------------
#define SCAN_B 256
__global__ void scan_local_kernel(const int* __restrict__ in, int* __restrict__ scanTmp,
                                  int* __restrict__ bsums, int N) {
    __shared__ int sh[SCAN_B];
    int t = threadIdx.x;
    int g = blockIdx.x * SCAN_B + t;
    sh[t] = (g < N) ? in[g] : 0;
    __syncthreads();
    for (int off = 1; off < SCAN_B; off <<= 1) {
        int add = (t >= off) ? sh[t - off] : 0;
        __syncthreads();
        sh[t] += add;
        __syncthreads();
    }
    if (g < N) scanTmp[g] = sh[t];              // inclusive within block
    if (t == SCAN_B - 1) bsums[blockIdx.x] = sh[t];
}

__global__ void scan_blocks_kernel(int* __restrict__ bsums, int nb) {
    __shared__ int sh[SCAN_B];
    int t = threadIdx.x;
    int orig = (t < nb) ? bsums[t] : 0;
    sh[t] = orig;
    __syncthreads();
    for (int off = 1; off < SCAN_B; off <<= 1) {
        int add = (t >= off) ? sh[t - off] : 0;
        __syncthreads();
        sh[t] += add;
        __syncthreads();
    }
    if (t < nb) bsums[t] = sh[t] - orig;        // exclusive block offsets
}

__global__ void scan_write_kernel(const int* __restrict__ scanTmp, const int* __restrict__ ideg,
                                  const int* __restrict__ bsums, int* __restrict__ rowptr,
                                  int N, int E) {
    int g = blockIdx.x * blockDim.x + threadIdx.x;
    if (g < N) rowptr[g] = scanTmp[g] - ideg[g] + bsums[g / SCAN_B];  // exclusive
    else if (g == N) rowptr[N] = E;
}

__global__ void bucket_kernel(const int* __restrict__ src, const int* __restrict__ dst,
                              const int* __restrict__ rowptr, int* __restrict__ cursor,
                              int* __restrict__ ssrc, int E) {
    int e = blockIdx.x * blockDim.x + threadIdx.x;
    if (e >= E) return;
    int d = dst[e];
    int pos = rowptr[d] + atomicAdd(&cursor[d], 1);
    ssrc[pos] = src[e];
}

// ---------------- CSR pull aggregation ----------------
// out[i] = inorm[i] * sum_{e in CSR(i)} x[src_e] * onorm[src_e]
// one wave32 per dst node. DIM multiple of 128: lane covers features 4*lane..4*lane+3
// per 128-chunk via a single b128 load (rows are 512B-aligned). DIM=16: scalar path.
template <int DIM>
__global__ void agg_csr_kernel(const float* __restrict__ x, const int* __restrict__ rowptr,
                               const int* __restrict__ ssrc, const float* __restrict__ onorm,
                               const float* __restrict__ inorm, float* __restrict__ out, int N) {
    int node = blockIdx.x * (blockDim.x >> 5) + (threadIdx.x >> 5);
    if (node >= N) return;
    int lane = threadIdx.x & 31;
    int beg = rowptr[node], end = rowptr[node + 1];
    float nw = inorm[node];

    if constexpr (DIM % 128 == 0) {
        constexpr int NV = DIM / 128;          // float4 chunks per lane
        v4f acc[NV];
#pragma unroll
        for (int j = 0; j < NV; ++j) acc[j] = (v4f){0.f, 0.f, 0.f, 0.f};
        for (int e = beg; e < end; ++e) {
            int s = ssrc[e];
            float w = onorm[s];
            const v4f* xr = (const v4f*)(x + (size_t)s * DIM);
#pragma unroll
            for (int j = 0; j < NV; ++j) {
                v4f v = xr[lane + 32 * j];
                acc[j] += v * w;
            }
        }
        v4f* orow = (v4f*)(out + (size_t)node * DIM);
#pragma unroll
        for (int j = 0; j < NV; ++j) orow[lane + 32 * j] = acc[j] * nw;
    } else {
        float acc = 0.0f;
        for (int e = beg; e < end; ++e) {
            int s = ssrc[e];
            float w = onorm[s];
            if (lane < DIM) acc += x[(size_t)s * DIM + lane] * w;
        }
        if (lane < DIM) out[(size_t)node * DIM + lane] = acc * nw;
    }
}

// ---------------- f32 WMMA GEMM: C[N x NOUT] = act(A[N x K] @ B[K x NOUT] + bias)
// one wave per 16 x (16*NT) strip; NT accumulators share a single A fragment per
// k-step (4x A reuse for NOUT=128). V_WMMA_F32_16X16X4_F32, 32 k-steps for K=128.
template <int K, int NOUT, bool RELU, bool BIAS>
__global__ void gemm_wmma_kernel(const float* __restrict__ A, const float* __restrict__ B,
                                 const float* __restrict__ bias, float* __restrict__ C,
                                 int rowTiles) {
    constexpr int NT = (NOUT % 64 == 0) ? 4 : 1;    // 16-col sub-tiles per wave
    constexpr int colGroups = NOUT / (16 * NT);
    int wg = blockIdx.x * 8 + (threadIdx.x >> 5);
    int rb = wg / colGroups;
    int cg = wg % colGroups;
    if (rb >= rowTiles) return;                    // wave-uniform exit (EXEC all-1s for WMMA)
    int lane = threadIdx.x & 31;
    int half = lane >> 4;                          // 0: K pair {0,1}, 1: K pair {2,3}
    int l16  = lane & 15;
    int arow = rb * 16 + l16;                      // A row for this lane
    int col0 = cg * 16 * NT + l16;                 // first B/D column for this lane
    const float* Ap = A + (size_t)arow * K + 2 * half;
    const float* Bp = B + (size_t)(2 * half) * NOUT + col0;
    v8f acc[NT];
#pragma unroll
    for (int j = 0; j < NT; ++j) acc[j] = (v8f){0.f, 0.f, 0.f, 0.f, 0.f, 0.f, 0.f, 0.f};
#pragma unroll 4
    for (int k4 = 0; k4 < K / 4; ++k4) {
        v2f a = *(const v2f*)(Ap + k4 * 4);        // A[arow][k4*4 + 2*half + {0,1}]
#pragma unroll
        for (int j = 0; j < NT; ++j) {
            v2f b;
            b.x = Bp[(k4 * 4 + 0) * NOUT + 16 * j];   // B[k4*4 + 2*half + 0][col0+16j]
            b.y = Bp[(k4 * 4 + 1) * NOUT + 16 * j];   // B[k4*4 + 2*half + 1][col0+16j]
            acc[j] = __builtin_amdgcn_wmma_f32_16x16x4_f32(false, a, false, b, (short)0,
                                                           acc[j], false, false);
        }
    }
#pragma unroll
    for (int j = 0; j < NT; ++j) {
        int col = col0 + 16 * j;
        float bv = BIAS ? bias[col] : 0.0f;
#pragma unroll
        for (int v = 0; v < 8; ++v) {
            int r = rb * 16 + v + 8 * half;        // D layout: VGPR v, half-wave selects M +8
            float val = acc[j][v] + bv;
            if (RELU) val = fmaxf(val, 0.0f);
            C[(size_t)r * NOUT + col] = val;
        }
    }
}

// scalar tail for rows not covered by 16-row tiles (N % 16 != 0 safety)
__global__ void gemm_tail_kernel(const float* __restrict__ A, const float* __restrict__ B,
                                 const float* __restrict__ bias, float* __restrict__ C,
                                 int rowStart, int N, int K, int NOUT, int relu, int hasBias) {
    int idx = blockIdx.x * blockDim.x + threadIdx.x;
    int rows = N - rowStart;
    if (idx >= rows * NOUT) return;
    int r = rowStart + idx / NOUT, c = idx % NOUT;
    float s = hasBias ? bias[c] : 0.0f;
    for (int k = 0; k < K; ++k) s += A[(size_t)r * K + k] * B[(size_t)k * NOUT + c];
    if (relu) s = fmaxf(s, 0.0f);
    C[(size_t)r * NOUT + c] = s;
}

// ---------------- per-graph mean pooling (LDS accumulate, then global atomics) ----------------
#define GG 64
#define CC 16
__global__ void pool_acc_kernel(const float* __restrict__ z, const int* __restrict__ gids,
                                float* __restrict__ accG, float* __restrict__ cnts, int N) {
    __shared__ float sAcc[GG * CC];
    __shared__ float sCnt[GG];
    for (int j = threadIdx.x; j < GG * CC; j += blockDim.x) sAcc[j] = 0.0f;
    for (int j = threadIdx.x; j < GG; j += blockDim.x) sCnt[j] = 0.0f;
    __syncthreads();
    int total = N * CC;
    for (int idx = blockIdx.x * blockDim.x + threadIdx.x; idx < total;
         idx += gridDim.x * blockDim.x) {
        int i = idx >> 4, c = idx & 15;
        int g = gids[i];
        atomicAdd(&sAcc[g * CC + c], z[idx]);
        if (c == 0) atomicAdd(&sCnt[g], 1.0f);
    }
    __syncthreads();
    for (int j = threadIdx.x; j < GG * CC; j += blockDim.x) atomicAdd(&accG[j], sAcc[j]);
    for (int j = threadIdx.x; j < GG; j += blockDim.x) atomicAdd(&cnts[j], sCnt[j]);
}

__global__ void pool_final_kernel(const float* __restrict__ accG, const float* __restrict__ cnts,
                                  const float* __restrict__ b4, float* __restrict__ out, int n) {
    int idx = blockIdx.x * blockDim.x + threadIdx.x;
    if (idx >= n) return;
    out[idx] = accG[idx] / fmaxf(cnts[idx >> 4], 1.0f) + b4[idx & 15];
}

// ---------------- host orchestration ----------------
extern "C" void kernel_launch(void* const* d_in, const int* in_sizes, int n_in,
                              void* d_out, int out_size, void* d_ws, size_t ws_size,
                              hipStream_t stream) {
    const int D = 128, H = 128;
    const float* in_feat = (const float*)d_in[0];
    const float* W1 = (const float*)d_in[1];  const float* b1 = (const float*)d_in[2];
    const float* W2 = (const float*)d_in[3];  const float* b2 = (const float*)d_in[4];
    const float* W3 = (const float*)d_in[5];  const float* b3 = (const float*)d_in[6];
    const float* W4 = (const float*)d_in[7];  const float* b4 = (const float*)d_in[8];
    const int* src  = (const int*)d_in[9];
    const int* dst  = (const int*)d_in[10];
    const int* gids = (const int*)d_in[11];
    const int N = in_sizes[0] / D;
    const int E = in_sizes[9];
    (void)n_in; (void)out_size; (void)ws_size;

    // workspace carve-out (256B aligned)
    char* w = (char*)d_ws;
    auto take = [&](size_t bytes) -> char* {
        char* p = w;
        w += (bytes + 255) & ~(size_t)255;
        return p;
    };
    float* onorm  = (float*)take((size_t)N * 4);
    float* inorm  = (float*)take((size_t)N * 4);
    int*   odeg   = (int*)take((size_t)N * 4);
    int*   ideg   = (int*)take((size_t)N * 4);
    int*   cursor = (int*)take((size_t)N * 4);
    int*   rowptr = (int*)take((size_t)(N + 1) * 4);
    int*   scanT  = (int*)take((size_t)N * 4);
    int*   bsums  = (int*)take((size_t)SCAN_B * 4);
    int*   ssrc   = (int*)take((size_t)E * 4);
    float* agg    = (float*)take((size_t)N * H * 4);
    float* hA     = (float*)take((size_t)N * H * 4);
    float* hB     = (float*)take((size_t)N * H * 4);
    float* y4     = (float*)take((size_t)N * CC * 4);
    float* z4     = (float*)take((size_t)N * CC * 4);
    float* accG   = (float*)take((size_t)GG * CC * 4);
    float* cnts   = (float*)take((size_t)GG * 4);

    const int tN = (N + 255) / 256;
    const int tE = (E + 255) / 256;

    // degrees + norms
    zero32_kernel<<<tN, 256, 0, stream>>>((unsigned int*)odeg, N);
    zero32_kernel<<<tN, 256, 0, stream>>>((unsigned int*)ideg, N);
    zero32_kernel<<<tN, 256, 0, stream>>>((unsigned int*)cursor, N);
    deg_kernel<<<tE, 256, 0, stream>>>(src, dst, odeg, ideg, E);
    norm_kernel<<<tN, 256, 0, stream>>>(odeg, ideg, onorm, inorm, N);

    // CSR by dst: rowptr = exscan(in_deg); bucket edges
    const int nb = (N + SCAN_B - 1) / SCAN_B;   // <= 256 for N <= 65536
    scan_local_kernel<<<nb, SCAN_B, 0, stream>>>(ideg, scanT, bsums, N);
    scan_blocks_kernel<<<1, SCAN_B, 0, stream>>>(bsums, nb);
    scan_write_kernel<<<(N + 1 + 255) / 256, 256, 0, stream>>>(scanT, ideg, bsums, rowptr, N, E);
    bucket_kernel<<<tE, 256, 0, stream>>>(src, dst, rowptr, cursor, ssrc, E);

    const int aggBlocks = (N + 7) / 8;          // 8 waves/block, one wave per node
    const int rowTiles  = N / 16;
    const int tail      = N - rowTiles * 16;
    const int gemmBlocksH = (rowTiles * 2 + 7) / 8;   // NOUT=128 -> 2 col groups of 64
    const int gemmBlocksC = (rowTiles + 7) / 8;       // NOUT=16  -> 1 col group of 16

    // layer 1: agg(in_feat) -> GEMM W1 + relu -> hA
    agg_csr_kernel<128><<<aggBlocks, 256, 0, stream>>>(in_feat, rowptr, ssrc, onorm, inorm, agg, N);
    gemm_wmma_kernel<128, 128, true, true><<<gemmBlocksH, 256, 0, stream>>>(agg, W1, b1, hA, rowTiles);
    if (tail > 0) gemm_tail_kernel<<<(tail * 128 + 255) / 256, 256, 0, stream>>>(agg, W1, b1, hA, rowTiles * 16, N, 128, 128, 1, 1);

    // layer 2
    agg_csr_kernel<128><<<aggBlocks, 256, 0, stream>>>(hA, rowptr, ssrc, onorm, inorm, agg, N);
    gemm_wmma_kernel<128, 128, true, true><<<gemmBlocksH, 256, 0, stream>>>(agg, W2, b2, hB, rowTiles);
    if (tail > 0) gemm_tail_kernel<<<(tail * 128 + 255) / 256, 256, 0, stream>>>(agg, W2, b2, hB, rowTiles * 16, N, 128, 128, 1, 1);

    // layer 3
    agg_csr_kernel<128><<<aggBlocks, 256, 0, stream>>>(hB, rowptr, ssrc, onorm, inorm, agg, N);
    gemm_wmma_kernel<128, 128, true, true><<<gemmBlocksH, 256, 0, stream>>>(agg, W3, b3, hA, rowTiles);
    if (tail > 0) gemm_tail_kernel<<<(tail * 128 + 255) / 256, 256, 0, stream>>>(agg, W3, b3, hA, rowTiles * 16, N, 128, 128, 1, 1);

    // layer 4: GEMM first (aggregation commutes with xW), 8x less edge traffic
    gemm_wmma_kernel<128, 16, false, false><<<gemmBlocksC, 256, 0, stream>>>(hA, W4, nullptr, y4, rowTiles);
    if (tail > 0) gemm_tail_kernel<<<(tail * 16 + 255) / 256, 256, 0, stream>>>(hA, W4, nullptr, y4, rowTiles * 16, N, 128, 16, 0, 0);
    agg_csr_kernel<16><<<aggBlocks, 256, 0, stream>>>(y4, rowptr, ssrc, onorm, inorm, z4, N);

    // per-graph mean pool + bias (bias of layer 4 commutes through the mean)
    zero32_kernel<<<(GG * CC + 255) / 256, 256, 0, stream>>>((unsigned int*)accG, GG * CC);
    zero32_kernel<<<1, 256, 0, stream>>>((unsigned int*)cnts, GG);
    pool_acc_kernel<<<256, 256, 0, stream>>>(z4, gids, accG, cnts, N);
    pool_final_kernel<<<(GG * CC + 255) / 256, 256, 0, stream>>>(accG, cnts, b4, (float*)d_out, GG * CC);
}